// FSIPM_55095840473455
// MI455X (gfx1250) — compile-verified
//
#include <hip/hip_runtime.h>
#include <math.h>

typedef __attribute__((ext_vector_type(2))) float v2f;
typedef __attribute__((ext_vector_type(8))) float v8f;

#define B_  64
#define R_  1024
#define T_  256
#define C1_ 8
#define C1P_ 16   // C1 padded to WMMA N=16
#define C2_ 16
#define PS_ 4
#define RP_ 256   // ROIS / PS pooled rows
#define H_  256   // ROIS / 4 hidden

// ---------------------------------------------------------------------------
// K1: per-row centering + inverse norm.  One wave32 per (b,r) row of T=256.
// ---------------------------------------------------------------------------
__global__ void k_center_norm(const float* __restrict__ x, float* __restrict__ xc,
                              float* __restrict__ invn) {
  int lane = threadIdx.x & 31;
  int widx = threadIdx.x >> 5;
  int row  = blockIdx.x * 8 + widx;            // [0, B*R)
  const float* xp = x + (size_t)row * T_;
  float v[8];
  float s = 0.f;
#pragma unroll
  for (int q = 0; q < 8; ++q) { v[q] = xp[lane + 32 * q]; s += v[q]; }
#pragma unroll
  for (int m = 16; m >= 1; m >>= 1) s += __shfl_xor(s, m, 32);
  float mean = s * (1.0f / T_);
  float ss = 0.f;
  float* xcp = xc + (size_t)row * T_;
#pragma unroll
  for (int q = 0; q < 8; ++q) { float d = v[q] - mean; xcp[lane + 32 * q] = d; ss += d * d; }
#pragma unroll
  for (int m = 16; m >= 1; m >>= 1) ss += __shfl_xor(ss, m, 32);
  if (lane == 0) invn[row] = rsqrtf(ss);
}

// ---------------------------------------------------------------------------
// K2: y[b,o,t] = sum_r (edge_w[o,r] * invn[b,r]) * xc[b,r,t].  Block per b.
//     y is stored padded to 16 rows per batch; rows 8..15 are zero so the
//     WMMA B operand can be loaded branch-free.
// ---------------------------------------------------------------------------
__global__ void k_edge_proj_y(const float* __restrict__ xc, const float* __restrict__ invn,
                              const float* __restrict__ edge_w, float* __restrict__ y) {
  __shared__ float wl[C1_ * R_];               // 32 KB of the 320 KB WGP LDS
  int b   = blockIdx.x;
  int tid = threadIdx.x;
  for (int k = 0; k < (C1_ * R_) / 256; ++k) {
    int i = tid + 256 * k;
    wl[i] = edge_w[i] * invn[b * R_ + (i & (R_ - 1))];
  }
  __syncthreads();
  float acc[C1_] = {};
  const float* xb = xc + (size_t)b * R_ * T_;
  for (int r = 0; r < R_; ++r) {
    if (((r & 63) == 0) && (r + 64 < R_))
      __builtin_prefetch(&xb[(r + 64) * T_ + tid], 0, 1);   // global_prefetch_b8
    float xv = xb[r * T_ + tid];
#pragma unroll
    for (int o = 0; o < C1_; ++o) acc[o] += wl[o * R_ + r] * xv;
  }
#pragma unroll
  for (int o = 0; o < C1_; ++o) y[(b * C1P_ + o) * T_ + tid] = acc[o];
#pragma unroll
  for (int o = C1_; o < C1P_; ++o) y[(b * C1P_ + o) * T_ + tid] = 0.f;  // zero pad rows
}

// ---------------------------------------------------------------------------
// K3: WMMA f32 16x16x4.  e[b,r,o] = leaky(invn[b,r]*<xc_r, y_o> + edge_b[o]);
//     a[b,r] = mean_o e.  One wave per 16-row tile; B loads are branch-free
//     because y is zero-padded to 16 rows.
// ---------------------------------------------------------------------------
__global__ void k_edge_wmma(const float* __restrict__ xc, const float* __restrict__ y,
                            const float* __restrict__ invn, const float* __restrict__ edge_b,
                            float* __restrict__ e, float* __restrict__ a) {
  int lane = threadIdx.x & 31;
  int widx = threadIdx.x >> 5;
  int wg   = blockIdx.x * 8 + widx;            // 4096 waves total
  int b    = wg >> 6;
  int r0   = (wg & 63) << 4;
  int hi   = lane >> 4;                        // K half: +0 or +2
  int l    = lane & 15;
  const float* arow = xc + ((size_t)(b * R_) + r0 + l) * T_;
  const float* brow = y + (size_t)(b * C1P_ + l) * T_;   // rows 8..15 are zeros
  v8f c = {};
  for (int kk = 0; kk < T_; kk += 4) {
    int K = kk + 2 * hi;
    v2f av, bv;
    av.x = arow[K]; av.y = arow[K + 1];
    bv.x = brow[K]; bv.y = brow[K + 1];
    c = __builtin_amdgcn_wmma_f32_16x16x4_f32(false, av, false, bv, (short)0, c, false, false);
  }
  bool cvalid = (l < C1_);
  float ebias = cvalid ? edge_b[l] : 0.f;
#pragma unroll
  for (int p = 0; p < 8; ++p) {
    int row = r0 + p + 8 * hi;                 // C/D layout: VGPR p -> M = p (+8 hi half)
    float v = c[p] * invn[b * R_ + row] + ebias;
    v = (v > 0.f) ? v : 0.01f * v;             // leaky_relu (cols >= 8 are exact zeros, unused)
    if (cvalid) e[((size_t)(b * R_) + row) * C1_ + l] = v;
    float s = v;                               // reduce over the 8 valid cols (lane group of 8)
    s += __shfl_xor(s, 1, 32);
    s += __shfl_xor(s, 2, 32);
    s += __shfl_xor(s, 4, 32);
    if ((lane & 15) == 0) a[b * R_ + row] = s * (1.0f / C1_);
  }
}

// ---------------------------------------------------------------------------
// K4: h = relu(a @ W1^T + b1).  Block per b, thread per hidden unit.
// ---------------------------------------------------------------------------
__global__ void k_mlp1(const float* __restrict__ a, const float* __restrict__ w1,
                       const float* __restrict__ b1, float* __restrict__ h) {
  __shared__ float al[R_];
  int b = blockIdx.x, tid = threadIdx.x;
  for (int q = 0; q < 4; ++q) al[tid + 256 * q] = a[b * R_ + tid + 256 * q];
  __syncthreads();
  float acc = b1[tid];
  const float* wr = w1 + (size_t)tid * R_;
  for (int r = 0; r < R_; ++r) acc += wr[r] * al[r];
  h[b * H_ + tid] = fmaxf(acc, 0.f);
}

// ---------------------------------------------------------------------------
// K5: att = sigmoid(h @ W2^T + b2).  Block per b.
// ---------------------------------------------------------------------------
__global__ void k_mlp2_att(const float* __restrict__ h, const float* __restrict__ w2,
                           const float* __restrict__ b2, float* __restrict__ att) {
  __shared__ float hl[H_];
  int b = blockIdx.x, tid = threadIdx.x;
  hl[tid] = h[b * H_ + tid];
  __syncthreads();
  for (int q = 0; q < 4; ++q) {
    int r = tid + 256 * q;
    float acc = b2[r];
    const float* wr = w2 + (size_t)r * H_;
    for (int j = 0; j < H_; ++j) acc += wr[j] * hl[j];
    att[b * R_ + r] = 1.f / (1.f + expf(-acc));
  }
}

// ---------------------------------------------------------------------------
// K6: thr[b] = 5th largest att[b,:] (PS=4 -> sorted[4]); 5 max extractions.
// ---------------------------------------------------------------------------
__global__ void k_thr(const float* __restrict__ att, float* __restrict__ thr) {
  __shared__ float vals[R_];
  __shared__ float rv[256];
  __shared__ int   ri[256];
  __shared__ int   smax;
  int b = blockIdx.x, tid = threadIdx.x;
  for (int q = 0; q < 4; ++q) vals[tid + 256 * q] = att[b * R_ + tid + 256 * q];
  __syncthreads();
  for (int it = 0; it <= PS_; ++it) {
    float lv = -1e30f; int li = R_;
    for (int q = 0; q < 4; ++q) {
      int r = tid + 256 * q;
      float v = vals[r];
      if (v > lv) { lv = v; li = r; }
    }
    rv[tid] = lv; ri[tid] = li;
    __syncthreads();
    for (int s = 128; s >= 1; s >>= 1) {
      if (tid < s) {
        if (rv[tid + s] > rv[tid] || (rv[tid + s] == rv[tid] && ri[tid + s] < ri[tid])) {
          rv[tid] = rv[tid + s]; ri[tid] = ri[tid + s];
        }
      }
      __syncthreads();
    }
    if (tid == 0) { smax = ri[0]; if (it == PS_) thr[b] = rv[0]; }
    __syncthreads();
    if (tid == 0) vals[smax] = -1e30f;
    __syncthreads();
  }
}

// ---------------------------------------------------------------------------
// K7: M[r] = sum_b (att[b,r] >= thr[b])
// ---------------------------------------------------------------------------
__global__ void k_mask_count(const float* __restrict__ att, const float* __restrict__ thr,
                             int* __restrict__ M) {
  int r = blockIdx.x * 256 + threadIdx.x;
  int c = 0;
  for (int b = 0; b < B_; ++b) c += (att[b * R_ + r] >= thr[b]) ? 1 : 0;
  M[r] = c;
}

// ---------------------------------------------------------------------------
// K8: stable top-256 of M (desc by M, asc by index) via exact rank counting.
//     key = (M<<10) | (1023 - r): strictly distinct -> unique ranks.
// ---------------------------------------------------------------------------
__global__ void k_topm(const int* __restrict__ M, int* __restrict__ topm,
                       float* __restrict__ topm_f) {
  __shared__ int keys[R_];
  int tid = threadIdx.x;
  int mykey = (M[tid] << 10) | (R_ - 1 - tid);
  keys[tid] = mykey;
  __syncthreads();
  int rank = 0;
  for (int i = 0; i < R_; ++i) rank += (keys[i] > mykey) ? 1 : 0;
  if (rank < RP_) { topm[rank] = tid; topm_f[rank] = (float)tid; }
}

// ---------------------------------------------------------------------------
// K9: node_feat[b,o] = leaky(sum_{r,c} att[b,r]*e[b,r,c]*node_w[o,c,r] + nb[o])
// ---------------------------------------------------------------------------
__global__ void k_node(const float* __restrict__ e, const float* __restrict__ att,
                       const float* __restrict__ nw, const float* __restrict__ nb,
                       float* __restrict__ out) {
  __shared__ float red[256];
  int b = blockIdx.x, tid = threadIdx.x;
  float acc[C2_] = {};
  for (int q = 0; q < 4; ++q) {
    int r = tid + 256 * q;
    float av = att[b * R_ + r];
    float sc[C1_];
#pragma unroll
    for (int cc = 0; cc < C1_; ++cc) sc[cc] = av * e[((size_t)(b * R_) + r) * C1_ + cc];
#pragma unroll
    for (int o = 0; o < C2_; ++o) {
      float t = 0.f;
#pragma unroll
      for (int cc = 0; cc < C1_; ++cc) t += sc[cc] * nw[(o * C1_ + cc) * R_ + r];
      acc[o] += t;
    }
  }
  for (int o = 0; o < C2_; ++o) {
    red[tid] = acc[o];
    __syncthreads();
    for (int s = 128; s >= 1; s >>= 1) { if (tid < s) red[tid] += red[tid + s]; __syncthreads(); }
    if (tid == 0) { float v = red[0] + nb[o]; out[b * C2_ + o] = (v > 0.f) ? v : 0.01f * v; }
    __syncthreads();
  }
}

// ---------------------------------------------------------------------------
// K10: x_pool gather: out[b,i,t] = x[b, topm[i], t]
// ---------------------------------------------------------------------------
__global__ void k_xpool(const float* __restrict__ x, const int* __restrict__ topm,
                        float* __restrict__ out) {
  int bi = blockIdx.x;                         // b*RP_ + i
  int b = bi >> 8, i = bi & 255;
  int r = topm[i];
  out[(size_t)bi * T_ + threadIdx.x] = x[((size_t)(b * R_) + r) * T_ + threadIdx.x];
}

// ---------------------------------------------------------------------------
// K11: WMMA f32 16x16x4 recompute of pooled adjacency:
//      adj[b,i,j] = invn_i*invn_j * <xc[b,topm[i]], xc[b,topm[j]]>.
//      One wave per 16x16 output tile, K-loop over T=256.
// ---------------------------------------------------------------------------
__global__ void k_adj_wmma(const float* __restrict__ xc, const float* __restrict__ invn,
                           const int* __restrict__ topm, float* __restrict__ adj) {
  int lane = threadIdx.x & 31;
  int widx = threadIdx.x >> 5;
  int wg   = blockIdx.x * 8 + widx;            // 16384 waves total
  int b    = wg >> 8;
  int tile = wg & 255;
  int i0 = (tile >> 4) << 4;
  int j0 = (tile & 15) << 4;
  int hi = lane >> 4;
  int l  = lane & 15;
  int rowA = topm[i0 + l];
  int rowB = topm[j0 + l];
  const float* ap = xc + ((size_t)(b * R_) + rowA) * T_;
  const float* bp = xc + ((size_t)(b * R_) + rowB) * T_;
  v8f c = {};
  for (int kk = 0; kk < T_; kk += 4) {
    int K = kk + 2 * hi;
    v2f av, bv;
    av.x = ap[K]; av.y = ap[K + 1];
    bv.x = bp[K]; bv.y = bp[K + 1];
    c = __builtin_amdgcn_wmma_f32_16x16x4_f32(false, av, false, bv, (short)0, c, false, false);
  }
  float invB = invn[b * R_ + rowB];
#pragma unroll
  for (int p = 0; p < 8; ++p) {
    int Mrow = p + 8 * hi;
    int mi = topm[i0 + Mrow];
    float invA = invn[b * R_ + mi];
    adj[((size_t)(b * RP_) + i0 + Mrow) * RP_ + j0 + l] = c[p] * invA * invB;
  }
}

// ---------------------------------------------------------------------------
extern "C" void kernel_launch(void* const* d_in, const int* in_sizes, int n_in,
                              void* d_out, int out_size, void* d_ws, size_t ws_size,
                              hipStream_t stream) {
  const float* x      = (const float*)d_in[0];
  const float* edge_w = (const float*)d_in[1];
  const float* edge_b = (const float*)d_in[2];
  const float* w1     = (const float*)d_in[3];
  const float* b1     = (const float*)d_in[4];
  const float* w2     = (const float*)d_in[5];
  const float* b2     = (const float*)d_in[6];
  const float* nw     = (const float*)d_in[7];
  const float* nb     = (const float*)d_in[8];
  float* out = (float*)d_out;

  // Workspace layout (floats), ~71 MB total.
  float* ws = (float*)d_ws;
  size_t off = 0;
  float* xc   = ws + off; off += (size_t)B_ * R_ * T_;    // centered x (64 MB)
  float* invn = ws + off; off += (size_t)B_ * R_;
  float* y    = ws + off; off += (size_t)B_ * C1P_ * T_;  // padded to 16 rows/batch
  float* e    = ws + off; off += (size_t)B_ * R_ * C1_;
  float* a    = ws + off; off += (size_t)B_ * R_;
  float* h    = ws + off; off += (size_t)B_ * H_;
  float* att  = ws + off; off += (size_t)B_ * R_;
  float* thr  = ws + off; off += (size_t)B_;
  int*   M    = (int*)(ws + off); off += R_;
  int*   topm = (int*)(ws + off); off += RP_;

  // Output layout: node_feat | x_pool | adj_pool | top_m (as float)
  float* out_node = out;                                   // 1024
  float* out_xp   = out + B_ * C2_;                        // 4,194,304
  float* out_adj  = out_xp + (size_t)B_ * RP_ * T_;        // 4,194,304
  float* out_topm = out_adj + (size_t)B_ * RP_ * RP_;      // 256

  k_center_norm<<<(B_ * R_) / 8, 256, 0, stream>>>(x, xc, invn);
  k_edge_proj_y<<<B_, 256, 0, stream>>>(xc, invn, edge_w, y);
  k_edge_wmma<<<512, 256, 0, stream>>>(xc, y, invn, edge_b, e, a);
  k_mlp1<<<B_, 256, 0, stream>>>(a, w1, b1, h);
  k_mlp2_att<<<B_, 256, 0, stream>>>(h, w2, b2, att);
  k_thr<<<B_, 256, 0, stream>>>(att, thr);
  k_mask_count<<<R_ / 256, 256, 0, stream>>>(att, thr, M);
  k_topm<<<1, R_, 0, stream>>>(M, topm, out_topm);
  k_node<<<B_, 256, 0, stream>>>(e, att, nw, nb, out_node);
  k_xpool<<<B_ * RP_, T_, 0, stream>>>(x, topm, out_xp);
  k_adj_wmma<<<2048, 256, 0, stream>>>(xc, invn, topm, out_adj);
}